// MLPDecoder_39487929319518
// MI455X (gfx1250) — compile-verified
//
#include <hip/hip_runtime.h>

#define N_PTS 2048
#define HD    128
#define ROWP  132   // LDS row stride in floats: keeps 16B alignment, bank-conflict free

typedef __attribute__((ext_vector_type(2))) float v2f;
typedef __attribute__((ext_vector_type(8))) float v8f;

// ---------------------------------------------------------------------------
// Kernel 1: fp32 WMMA GEMM precompute.
//   z=0: ws[0      ..] A[i][k] = sum_m x[i][m] * W1[m][k]
//   z=1: ws[N*H    ..] B[j][k] = sum_m x[j][m] * W1[128+m][k] + b1[k]
// One wave32 per 16x16 output tile, V_WMMA_F32_16X16X4_F32, K-loop over 128.
// ---------------------------------------------------------------------------
__global__ __launch_bounds__(32) void precompute_gemm(
    const float* __restrict__ x, const float* __restrict__ W1,
    const float* __restrict__ b1, float* __restrict__ ws)
{
    const int lane = threadIdx.x;
    const int lr   = lane & 15;   // M (A) / N (B,D) index within half
    const int half = lane >> 4;   // K-pair select for A/B, +8 rows for D
    const int k0 = blockIdx.x * 16;
    const int i0 = blockIdx.y * 16;
    const int z  = blockIdx.z;

    const float* W  = W1 + (size_t)z * HD * HD;   // top / bottom half of W1
    float* Out      = ws + (size_t)z * N_PTS * HD;

    v8f d = {0.f, 0.f, 0.f, 0.f, 0.f, 0.f, 0.f, 0.f};

    const float* xrow = x + (size_t)(i0 + lr) * HD;
    #pragma unroll 4
    for (int m = 0; m < HD; m += 4) {
        const int ma = m + 2 * half;
        // A 16x4: VGPR0 = K={0|2}, VGPR1 = K={1|3}; contiguous in memory -> b64 load
        v2f a = *(const v2f*)(xrow + ma);
        // B 4x16: row striped across lanes within a VGPR (coalesced across lr)
        v2f b;
        b[0] = W[(size_t)ma       * HD + k0 + lr];
        b[1] = W[(size_t)(ma + 1) * HD + k0 + lr];
        d = __builtin_amdgcn_wmma_f32_16x16x4_f32(
                /*neg_a=*/false, a, /*neg_b=*/false, b,
                /*c_mod=*/(short)0, d, /*reuse_a=*/false, /*reuse_b=*/false);
    }

    const float bias = z ? b1[k0 + lr] : 0.0f;
    #pragma unroll
    for (int r = 0; r < 8; ++r) {
        Out[(size_t)(i0 + r + 8 * half) * HD + k0 + lr] = d[r] + bias;
    }
}

// ---------------------------------------------------------------------------
// Kernel 2: pairwise ELU-MLP over the upper triangle, 16x16 pair tiles.
//   e(i,j) = sum_k elu(A[i,k] + B[j,k]) * W2[k] + b2,  adj[i,j]=adj[j,i]=e
// 256 threads = one (i,j) pair each; A/B tiles + W2 staged in LDS; e-tile
// staged in LDS so both the tile and its transpose are written coalesced.
// ---------------------------------------------------------------------------
__global__ __launch_bounds__(256) void pair_mlp(
    const float* __restrict__ ws, const float* __restrict__ W2,
    const float* __restrict__ b2, float* __restrict__ adj)
{
    const int it = blockIdx.y;
    const int jt = blockIdx.x;
    if (jt < it) return;   // upper triangle only (uniform exit)

    __shared__ float sA[16 * ROWP];
    __shared__ float sB[16 * ROWP];
    __shared__ float sW2[HD];
    __shared__ float sE[16 * 17];

    const int tid = threadIdx.x;
    const float* Afull = ws;
    const float* Bfull = ws + (size_t)N_PTS * HD;
    const int i0 = it * 16;
    const int j0 = jt * 16;

    // Stage A/B tiles: 16 rows x 128 floats each = 512 float4s, 2 per thread.
    #pragma unroll
    for (int v = 0; v < 2; ++v) {
        const int flat = tid + v * 256;       // float4 index
        const int r = flat >> 5;              // row 0..15
        const int c = (flat & 31) << 2;       // col 0..124
        *(float4*)(sA + r * ROWP + c) = *(const float4*)(Afull + (size_t)(i0 + r) * HD + c);
        *(float4*)(sB + r * ROWP + c) = *(const float4*)(Bfull + (size_t)(j0 + r) * HD + c);
    }
    if (tid < HD) sW2[tid] = W2[tid];
    __syncthreads();

    const int tj = tid & 15;   // j within tile
    const int ti = tid >> 4;   // i within tile
    const float* Ar = sA + ti * ROWP;
    const float* Br = sB + tj * ROWP;

    float acc = 0.0f;
    #pragma unroll 8
    for (int k = 0; k < HD; k += 4) {
        const float4 a4 = *(const float4*)(Ar + k);
        const float4 b4 = *(const float4*)(Br + k);
        const float4 w4 = *(const float4*)(sW2 + k);
        float s0 = a4.x + b4.x; float t0 = s0 > 0.f ? s0 : (__expf(s0) - 1.f);
        float s1 = a4.y + b4.y; float t1 = s1 > 0.f ? s1 : (__expf(s1) - 1.f);
        float s2 = a4.z + b4.z; float t2 = s2 > 0.f ? s2 : (__expf(s2) - 1.f);
        float s3 = a4.w + b4.w; float t3 = s3 > 0.f ? s3 : (__expf(s3) - 1.f);
        acc = fmaf(t0, w4.x, acc);
        acc = fmaf(t1, w4.y, acc);
        acc = fmaf(t2, w4.z, acc);
        acc = fmaf(t3, w4.w, acc);
    }
    const float ev = acc + b2[0];
    sE[ti * 17 + tj] = ev;
    __syncthreads();

    // Coalesced writes: u = fast (column) index, v = row index.
    const int u = tid & 15;
    const int v = tid >> 4;
    if (it == jt) {
        const float val = (v < u) ? sE[v * 17 + u]
                        : (v > u) ? sE[u * 17 + v] : 0.0f;
        adj[(size_t)(i0 + v) * N_PTS + j0 + u] = val;
    } else {
        adj[(size_t)(i0 + v) * N_PTS + j0 + u] = sE[v * 17 + u];  // upper tile
        adj[(size_t)(j0 + v) * N_PTS + i0 + u] = sE[u * 17 + v];  // mirrored tile
    }
}

// ---------------------------------------------------------------------------
extern "C" void kernel_launch(void* const* d_in, const int* in_sizes, int n_in,
                              void* d_out, int out_size, void* d_ws, size_t ws_size,
                              hipStream_t stream) {
    const float* x  = (const float*)d_in[0];   // [N, H]
    const float* W1 = (const float*)d_in[1];   // [2H, H]
    const float* b1 = (const float*)d_in[2];   // [H]
    const float* W2 = (const float*)d_in[3];   // [H, 1]
    const float* b2 = (const float*)d_in[4];   // [1]
    float* out = (float*)d_out;                // [N, N]
    float* ws  = (float*)d_ws;                 // needs 2*N*H floats = 2 MB

    // Kernel 1: A and B precompute via fp32 WMMA (grid: k-tiles x i-tiles x {top,bot})
    dim3 g1(HD / 16, N_PTS / 16, 2);
    precompute_gemm<<<g1, 32, 0, stream>>>(x, W1, b1, ws);

    // Kernel 2: pairwise tiles over the upper triangle (lower-tri blocks exit)
    dim3 g2(N_PTS / 16, N_PTS / 16);
    pair_mlp<<<g2, 256, 0, stream>>>(ws, W2, b2, out);
}